// Prediction_68676527063502
// MI455X (gfx1250) — compile-verified
//
#include <hip/hip_runtime.h>

typedef __attribute__((ext_vector_type(2))) float v2f;
typedef __attribute__((ext_vector_type(8))) float v8f;
typedef __attribute__((ext_vector_type(4))) float f4;

#define BDIM 16
#define LDIM 4096
#define BANDW 15

// ---------------------------------------------------------------------------
// Kernel 1: stream 1 GiB of zeros with non-temporal b128 stores.
// Store-bandwidth bound: ~46 us floor at 23.3 TB/s HBM.
// ---------------------------------------------------------------------------
__global__ void band_zero_fill(f4* __restrict__ out, unsigned long long nvec) {
  const unsigned long long stride = (unsigned long long)gridDim.x * blockDim.x;
  f4 z = {0.0f, 0.0f, 0.0f, 0.0f};
  for (unsigned long long v =
           (unsigned long long)blockIdx.x * blockDim.x + threadIdx.x;
       v < nvec; v += stride) {
    __builtin_nontemporal_store(z, out + v);
  }
}

// ---------------------------------------------------------------------------
// Kernel 2: compute the band via rank-1 WMMA outer products.
// One wave32 handles one 16-row tile of one batch: two 16x16 tiles
// (diagonal + super-diagonal) cover all nonzeros for those rows.
// A = [s_col | 0 0 0] (16x4 f32), B = [e_row ; 0 0 0] (4x16 f32)
// => D = s ⊗ e exactly (single f32 multiply, zero-adds exact).
// ---------------------------------------------------------------------------
__global__ void band_outer_wmma(const float* __restrict__ s,
                                const float* __restrict__ e,
                                float* __restrict__ out) {
  const int lane = threadIdx.x & 31;
  const int waveInBlock = threadIdx.x >> 5;
  const int tile = blockIdx.x * (blockDim.x >> 5) + waveInBlock;

  const int tilesPerBatch = LDIM / 16;          // 256
  const int b  = tile / tilesPerBatch;
  const int i0 = (tile - b * tilesPerBatch) * 16;

  const int  nl = lane & 15;
  const bool lo = lane < 16;
  const int  mBase = lo ? 0 : 8;                // C/D layout: VGPR r -> M=r (+8 upper half)

  // A-matrix 16x4 f32 layout: VGPR0 = K=0 (lanes 0-15) / K=2 (lanes 16-31),
  // VGPR1 = K=1 / K=3.  Only K=0 column is nonzero: s[b, i0+M].
  const float sv = s[(unsigned long long)b * LDIM + i0 + nl];
  v2f a;
  a.x = lo ? sv : 0.0f;
  a.y = 0.0f;

  // ---- Tile 0: diagonal block, j0 = i0 (covers 0 <= d <= 15 upper triangle)
  {
    const int j0 = i0;
    const float ev = e[(unsigned long long)b * LDIM + j0 + nl];
    v2f bm;
    bm.x = lo ? ev : 0.0f;   // B row K=0 striped across lanes 0-15
    bm.y = 0.0f;
    v8f c = {};
    v8f d = __builtin_amdgcn_wmma_f32_16x16x4_f32(
        /*neg_a=*/false, a, /*neg_b=*/false, bm,
        /*c_mod=*/(short)0, c, /*reuse_a=*/false, /*reuse_b=*/false);
#pragma unroll
    for (int r = 0; r < 8; ++r) {
      const int m = r + mBase;
      const int i = i0 + m;
      const int j = j0 + nl;
      const int dd = j - i;
      const float val = (dd >= 0 && dd <= BANDW) ? d[r] : 0.0f;
      out[((unsigned long long)b * LDIM + i) * LDIM + j] = val;
    }
  }

  // ---- Tile 1: super-diagonal block, j0 = i0 + 16 (strict lower triangle of
  // that tile carries d = 16 + n - m in [1,15]).  Skipped (wave-uniformly)
  // only for the final row tile where it would run past column L.
  const int j0 = i0 + 16;
  if (j0 < LDIM) {
    const float ev = e[(unsigned long long)b * LDIM + j0 + nl];
    v2f bm;
    bm.x = lo ? ev : 0.0f;
    bm.y = 0.0f;
    v8f c = {};
    v8f d = __builtin_amdgcn_wmma_f32_16x16x4_f32(
        false, a, false, bm, (short)0, c, false, false);
#pragma unroll
    for (int r = 0; r < 8; ++r) {
      const int m = r + mBase;
      const int i = i0 + m;
      const int j = j0 + nl;
      const int dd = j - i;
      const float val = (dd >= 0 && dd <= BANDW) ? d[r] : 0.0f;
      out[((unsigned long long)b * LDIM + i) * LDIM + j] = val;
    }
  }
}

// ---------------------------------------------------------------------------
extern "C" void kernel_launch(void* const* d_in, const int* in_sizes, int n_in,
                              void* d_out, int out_size, void* d_ws,
                              size_t ws_size, hipStream_t stream) {
  const float* s = (const float*)d_in[0];
  const float* e = (const float*)d_in[1];
  float* out = (float*)d_out;

  // 1) Zero the full [B, L, L] output: 2^26 float4 stores.
  const unsigned long long nvec =
      (unsigned long long)BDIM * LDIM * LDIM / 4;   // 67,108,864
  band_zero_fill<<<4096, 256, 0, stream>>>((f4*)out, nvec);

  // 2) Band tiles: 16 batches * 256 row-tiles = 4096 waves; 8 waves/block.
  const int tiles = BDIM * (LDIM / 16);             // 4096
  const int blocks = tiles / 8;                     // 512
  band_outer_wmma<<<blocks, 256, 0, stream>>>(s, e, out);
}